// gaussian_conv_pooling_14345190768658
// MI455X (gfx1250) — compile-verified
//
#include <hip/hip_runtime.h>
#include <hip/hip_bf16.h>
#include <math.h>

// ---------------------------------------------------------------------------
// Problem constants (from reference)
// ---------------------------------------------------------------------------
#define NK    12
#define NB    2
#define NC    256
#define NH    64
#define NW    64
#define NROI  256          // B*R
#define POOL  7
#define GRD   8
#define INDIM 12544        // 256*7*7
#define HID   4096
#define FS    7
#define PAD   3
#define TILE  70           // 64 + 2*3

// GEMM1 tiling
#define KC       64        // K-chunk staged in LDS
#define NCOLS    64        // columns per block
#define BSTRIDE  68        // padded LDS row stride (floats): (4c+k)%64 distinct,
                           // 16B-aligned for float4 staging stores

typedef __attribute__((ext_vector_type(2))) float v2f;
typedef __attribute__((ext_vector_type(8))) float v8f;

// ---------------------------------------------------------------------------
// K0: depthwise 7x7 gaussian conv, zero-padded.  One block per (k,b,c) plane.
// filtered layout: [k][b][c][64][64]
// ---------------------------------------------------------------------------
__global__ void k0_dwconv(const float* __restrict__ feat,
                          const float* __restrict__ kern,
                          float* __restrict__ filtered) {
  __shared__ float tile[TILE * TILE];
  __shared__ float kw[FS * FS];
  const int bid = blockIdx.x;                  // = k*512 + b*256 + c
  const int c   = bid & 255;
  const int b   = (bid >> 8) & 1;
  const int k   = bid >> 9;
  const int tid = threadIdx.x;

  const float* plane = feat + ((size_t)(b * NC + c) << 12);
  if (tid < FS * FS) kw[tid] = kern[(size_t)k * NC * FS * FS + tid];  // c==0 slice

  for (int i = tid; i < TILE * TILE; i += 256) {
    int ty = i / TILE, tx = i - ty * TILE;
    int y = ty - PAD, x = tx - PAD;
    float v = 0.0f;
    if ((unsigned)y < (unsigned)NH && (unsigned)x < (unsigned)NW)
      v = plane[(y << 6) + x];
    tile[i] = v;
  }
  __syncthreads();

  float* outp = filtered + ((size_t)bid << 12);
  for (int i = tid; i < NH * NW; i += 256) {
    int y = i >> 6, x = i & 63;
    float s = 0.0f;
#pragma unroll
    for (int dy = 0; dy < FS; ++dy)
#pragma unroll
      for (int dx = 0; dx < FS; ++dx)
        s = fmaf(tile[(y + dy) * TILE + (x + dx)], kw[dy * FS + dx], s);
    outp[i] = s;
  }
}

// ---------------------------------------------------------------------------
// Shared helper: per-ROI bilinear sample coordinates into LDS
// ---------------------------------------------------------------------------
__device__ __forceinline__ void roi_coords(const float* __restrict__ rois, int n,
                                           int* sx0, int* sx1, float* swx,
                                           int* sy0, int* sy1, float* swy,
                                           int* sb) {
  const int tid = threadIdx.x;
  if (tid == 0) *sb = (int)rois[n * 5 + 0];
  if (tid < GRD) {
    float x1 = rois[n * 5 + 1] * 0.0625f, x2 = rois[n * 5 + 3] * 0.0625f;
    float xs = x1 + (x2 - x1) * ((float)tid * (1.0f / 7.0f));
    xs = fminf(fmaxf(xs, 0.0f), 63.0f);
    float xf = floorf(xs);
    int xi = (int)xf;
    sx0[tid] = xi;
    sx1[tid] = min(xi + 1, 63);
    swx[tid] = xs - xf;
  } else if (tid < 2 * GRD) {
    int t = tid - GRD;
    float y1 = rois[n * 5 + 2] * 0.0625f, y2 = rois[n * 5 + 4] * 0.0625f;
    float ys = y1 + (y2 - y1) * ((float)t * (1.0f / 7.0f));
    ys = fminf(fmaxf(ys, 0.0f), 63.0f);
    float yf = floorf(ys);
    int yi = (int)yf;
    sy0[t] = yi << 6;                 // pre-scaled row offsets
    sy1[t] = min(yi + 1, 63) << 6;
    swy[t] = ys - yf;
  }
}

// ---------------------------------------------------------------------------
// K1: RoIAlignAvg of base_feat -> X matrix [NROI][INDIM] (row-major).
// One block per ROI, one thread per channel.
// ---------------------------------------------------------------------------
__global__ void k1_roialign_base(const float* __restrict__ feat,
                                 const float* __restrict__ rois,
                                 float* __restrict__ X) {
  __shared__ int sx0[GRD], sx1[GRD], sy0[GRD], sy1[GRD], sb;
  __shared__ float swx[GRD], swy[GRD];
  const int n = blockIdx.x;
  roi_coords(rois, n, sx0, sx1, swx, sy0, sy1, swy, &sb);
  __syncthreads();

  const int c = threadIdx.x;
  const float* plane = feat + ((size_t)(sb * NC + c) << 12);
  float* xo = X + (size_t)n * INDIM + (size_t)c * (POOL * POOL);

  float prev[GRD];
  for (int r = 0; r < GRD; ++r) {
    float cur[GRD];
    const int y0 = sy0[r], y1 = sy1[r];
    const float wy = swy[r];
#pragma unroll
    for (int j = 0; j < GRD; ++j) {
      float t0 = plane[y0 + sx0[j]], t1 = plane[y0 + sx1[j]];
      float b0 = plane[y1 + sx0[j]], b1 = plane[y1 + sx1[j]];
      float top = t0 + (t1 - t0) * swx[j];
      float bot = b0 + (b1 - b0) * swx[j];
      cur[j] = top + (bot - top) * wy;
    }
    if (r > 0) {
#pragma unroll
      for (int j = 0; j < POOL; ++j)
        xo[(r - 1) * POOL + j] =
            0.25f * (prev[j] + prev[j + 1] + cur[j] + cur[j + 1]);
    }
#pragma unroll
    for (int j = 0; j < GRD; ++j) prev[j] = cur[j];
  }
}

// ---------------------------------------------------------------------------
// K2: fused GEMM1 (X @ W1^T) + bias + BatchNorm + ReLU with
// V_WMMA_F32_16X16X4_F32 and LDS-staged B.
//
// Block tile: 128 rows x 64 cols.  grid.x = (HID/64)*2 = 128 blocks.
//   blockIdx.x & 1   -> which 128-row half of X
//   blockIdx.x >> 1  -> 64-column strip of HID
// 8 waves; wave w owns M-tile rows [mbase, mbase+16) and all 4 N-tiles,
// sharing one A fragment across 4 WMMAs per k-step (B from LDS).
// W1 hits HBM exactly once; L2 traffic ~1.2 GB vs ~5 GB unstaged.
// ---------------------------------------------------------------------------
__global__ void k2_gemm1(const float* __restrict__ X,
                         const float* __restrict__ W1,
                         const float* __restrict__ b1,
                         const float* __restrict__ g1,
                         const float* __restrict__ be1,
                         const float* __restrict__ m1,
                         const float* __restrict__ v1,
                         float* __restrict__ H) {
  __shared__ float bs[NCOLS * BSTRIDE];        // 64*68*4 = 17.4 KB

  const int tid  = threadIdx.x;
  const int lane = tid & 31;
  const int wave = tid >> 5;
  const int hf   = lane >> 4;                  // K-half of fragment
  const int l15  = lane & 15;

  const int mbase   = (blockIdx.x & 1) * 128 + (wave << 4);
  const int colbase = (blockIdx.x >> 1) << 6;

  // cooperative B-staging ids: 256 threads cover 64 cols x 4 k-quarters
  const int lc = tid & 63;                     // column within strip
  const int lk = (tid >> 6) << 4;              // k-offset within chunk

  const float* ap   = X + (size_t)(mbase + l15) * INDIM + 2 * hf;
  const float* wrow = W1 + (size_t)(colbase + lc) * INDIM + lk;
  float*       bsw  = &bs[lc * BSTRIDE + lk];

  v8f c[4] = {};

  for (int kc = 0; kc < INDIM; kc += KC) {
    __syncthreads();
    // stage W1[colbase..+63][kc..kc+63] into LDS (16 floats / thread)
#pragma unroll
    for (int i = 0; i < 4; ++i) {
      *(float4*)(bsw + i * 4) = *(const float4*)(wrow + kc + i * 4);
    }
    __builtin_prefetch((const void*)(wrow + kc + KC), 0, 1);  // next chunk
    __syncthreads();

#pragma unroll 4
    for (int kk = 0; kk < KC; kk += 4) {
      v2f a = *(const v2f*)(ap + kc + kk);
#pragma unroll
      for (int nt = 0; nt < 4; ++nt) {
        v2f bb = *(const v2f*)(&bs[(nt * 16 + l15) * BSTRIDE + kk + 2 * hf]);
        c[nt] = __builtin_amdgcn_wmma_f32_16x16x4_f32(false, a, false, bb,
                                                      (short)0, c[nt],
                                                      false, false);
      }
    }
  }

  // BN folded: h = acc*scale + shift, relu.  C layout: row = mbase + 8*hf + j.
#pragma unroll
  for (int nt = 0; nt < 4; ++nt) {
    const int col = colbase + (nt << 4) + l15;
    const float scale = g1[col] * rsqrtf(v1[col] + 1e-5f);
    const float shift = be1[col] + (b1[col] - m1[col]) * scale;
#pragma unroll
    for (int j = 0; j < 8; ++j) {
      const int r = mbase + (hf << 3) + j;
      H[(size_t)r * HID + col] = fmaxf(c[nt][j] * scale + shift, 0.0f);
    }
  }
}

// ---------------------------------------------------------------------------
// K3: GEMM2 (H @ W2^T) + BN + softmax over the 12 kernels.
// One block per sample row; LDS float atomics for the reduction.
// wgt layout: [12][NROI]
// ---------------------------------------------------------------------------
__global__ void k3_gemm2_softmax(const float* __restrict__ H,
                                 const float* __restrict__ W2,
                                 const float* __restrict__ b2,
                                 const float* __restrict__ g2,
                                 const float* __restrict__ be2,
                                 const float* __restrict__ m2,
                                 const float* __restrict__ v2,
                                 float* __restrict__ wgt) {
  __shared__ float sacc[NK];
  const int n = blockIdx.x, tid = threadIdx.x;
  if (tid < NK) sacc[tid] = 0.0f;
  __syncthreads();

  float acc[NK];
#pragma unroll
  for (int j = 0; j < NK; ++j) acc[j] = 0.0f;

  const float* hrow = H + (size_t)n * HID;
  for (int k = tid; k < HID; k += 256) {
    float hv = hrow[k];
#pragma unroll
    for (int j = 0; j < NK; ++j) acc[j] = fmaf(hv, W2[j * HID + k], acc[j]);
  }
#pragma unroll
  for (int j = 0; j < NK; ++j) atomicAdd(&sacc[j], acc[j]);
  __syncthreads();

  if (tid == 0) {
    float o[NK], mx = -1e30f;
#pragma unroll
    for (int j = 0; j < NK; ++j) {
      float s = g2[j] * rsqrtf(v2[j] + 1e-5f);
      o[j] = (sacc[j] + b2[j] - m2[j]) * s + be2[j];
      mx = fmaxf(mx, o[j]);
    }
    float sum = 0.0f;
#pragma unroll
    for (int j = 0; j < NK; ++j) { o[j] = __expf(o[j] - mx); sum += o[j]; }
    float inv = 1.0f / sum;
#pragma unroll
    for (int j = 0; j < NK; ++j) wgt[j * NROI + n] = o[j] * inv;
  }
}

// ---------------------------------------------------------------------------
// K4: fused RoIAlign of the 12 filtered maps + softmax-weighted sum.
// Avoids materializing pooled_list (154 MB).  One block per ROI, thread = ch.
// out: [NROI][NC][7][7]
// ---------------------------------------------------------------------------
__global__ void k4_weighted_pool(const float* __restrict__ filtered,
                                 const float* __restrict__ rois,
                                 const float* __restrict__ wgt,
                                 float* __restrict__ out) {
  __shared__ int sx0[GRD], sx1[GRD], sy0[GRD], sy1[GRD], sb;
  __shared__ float swx[GRD], swy[GRD], sw[NK];
  const int n = blockIdx.x;
  const int tid = threadIdx.x;
  roi_coords(rois, n, sx0, sx1, swx, sy0, sy1, swy, &sb);
  if (tid >= 16 && tid < 16 + NK) sw[tid - 16] = wgt[(tid - 16) * NROI + n];
  __syncthreads();

  const int c = tid;
  float acc[POOL * POOL];
#pragma unroll
  for (int i = 0; i < POOL * POOL; ++i) acc[i] = 0.0f;

  for (int k = 0; k < NK; ++k) {
    const float* plane =
        filtered + ((size_t)(k * (NB * NC) + sb * NC + c) << 12);
    const float wk = sw[k];
    float prev[GRD];
    for (int r = 0; r < GRD; ++r) {
      float cur[GRD];
      const int y0 = sy0[r], y1 = sy1[r];
      const float wy = swy[r];
#pragma unroll
      for (int j = 0; j < GRD; ++j) {
        float t0 = plane[y0 + sx0[j]], t1 = plane[y0 + sx1[j]];
        float b0 = plane[y1 + sx0[j]], b1 = plane[y1 + sx1[j]];
        float top = t0 + (t1 - t0) * swx[j];
        float bot = b0 + (b1 - b0) * swx[j];
        cur[j] = top + (bot - top) * wy;
      }
      if (r > 0) {
#pragma unroll
        for (int j = 0; j < POOL; ++j)
          acc[(r - 1) * POOL + j] = fmaf(
              wk, 0.25f * (prev[j] + prev[j + 1] + cur[j] + cur[j + 1]),
              acc[(r - 1) * POOL + j]);
      }
#pragma unroll
      for (int j = 0; j < GRD; ++j) prev[j] = cur[j];
    }
  }

  float* op = out + ((size_t)n * NC + c) * (POOL * POOL);
#pragma unroll
  for (int i = 0; i < POOL * POOL; ++i) op[i] = acc[i];
}

// ---------------------------------------------------------------------------
// launch
// ---------------------------------------------------------------------------
extern "C" void kernel_launch(void* const* d_in, const int* in_sizes, int n_in,
                              void* d_out, int out_size, void* d_ws,
                              size_t ws_size, hipStream_t stream) {
  const float* base_feat = (const float*)d_in[0];
  const float* kernel    = (const float*)d_in[1];
  const float* rois      = (const float*)d_in[2];
  const float* W1        = (const float*)d_in[3];
  const float* b1        = (const float*)d_in[4];
  const float* g1        = (const float*)d_in[5];
  const float* be1       = (const float*)d_in[6];
  const float* m1        = (const float*)d_in[7];
  const float* v1        = (const float*)d_in[8];
  const float* W2        = (const float*)d_in[9];
  const float* b2        = (const float*)d_in[10];
  const float* g2        = (const float*)d_in[11];
  const float* be2       = (const float*)d_in[12];
  const float* m2        = (const float*)d_in[13];
  const float* v2        = (const float*)d_in[14];
  float* outp = (float*)d_out;

  // workspace partition (floats)
  float* ws       = (float*)d_ws;
  float* filtered = ws;                                         // 12*2*256*4096
  float* X        = filtered + (size_t)NK * NB * NC * NH * NW;  // 256*12544
  float* H        = X + (size_t)NROI * INDIM;                   // 256*4096
  float* WGT      = H + (size_t)NROI * HID;                     // 12*256

  k0_dwconv<<<NK * NB * NC, 256, 0, stream>>>(base_feat, kernel, filtered);
  k1_roialign_base<<<NROI, 256, 0, stream>>>(base_feat, rois, X);
  k2_gemm1<<<(HID / NCOLS) * 2, 256, 0, stream>>>(X, W1, b1, g1, be1, m1, v1, H);
  k3_gemm2_softmax<<<NROI, 256, 0, stream>>>(H, W2, b2, g2, be2, m2, v2, WGT);
  k4_weighted_pool<<<NROI, 256, 0, stream>>>(filtered, rois, WGT, outp);
}